// GraphSAGE_46145128628312
// MI455X (gfx1250) — compile-verified
//
#include <hip/hip_runtime.h>
#include <hip/hip_bf16.h>

typedef float v2f __attribute__((ext_vector_type(2)));
typedef float v8f __attribute__((ext_vector_type(8)));

#define FEAT 128          // per-layer input width (D == H == 128)
#define KDIM 256          // concatenated K: [agg/deg | x]
#define COLSB 64          // output columns per block
#define NT (COLSB / 16)   // 4 column tiles of 16

// ---------------------------------------------------------------------------
// Utility: zero a float region (grid-stride).
// ---------------------------------------------------------------------------
__global__ void k_zero(float* __restrict__ p, long n) {
  long i = (long)blockIdx.x * blockDim.x + threadIdx.x;
  long stride = (long)gridDim.x * blockDim.x;
  for (; i < n; i += stride) p[i] = 0.0f;
}

// ---------------------------------------------------------------------------
// Degree: one thread per edge, float atomic add (resolves in L2).
// ---------------------------------------------------------------------------
__global__ void k_deg(const int* __restrict__ dst, float* __restrict__ deg, int E) {
  int e = blockIdx.x * blockDim.x + threadIdx.x;
  if (e < E) atomicAdd(&deg[dst[e]], 1.0f);
}

// ---------------------------------------------------------------------------
// Edge scatter-add: one wave per edge; 32 lanes x float4 covers a 128-float
// row (512B coalesced gather), then 4 global f32 atomics per lane into agg.
// ---------------------------------------------------------------------------
__global__ __launch_bounds__(256) void k_scatter(
    const float* __restrict__ feat, const int* __restrict__ src,
    const int* __restrict__ dst, float* __restrict__ agg, int E) {
  int gt = blockIdx.x * blockDim.x + threadIdx.x;
  int e = gt >> 5;
  int lane = gt & 31;
  if (e >= E) return;
  int s = src[e];
  int d = dst[e];
  float4 v = *(const float4*)(feat + (size_t)s * FEAT + lane * 4);
  float* p = agg + (size_t)d * FEAT + lane * 4;
  atomicAdd(p + 0, v.x);
  atomicAdd(p + 1, v.y);
  atomicAdd(p + 2, v.z);
  atomicAdd(p + 3, v.w);
}

// ---------------------------------------------------------------------------
// Fused SAGE GEMM:  out[16-node tile, 64 cols] = [agg/deg | x] @ [Wl ; Wr] + b
// One wave per 16-node tile; blockIdx.y selects which 64-column slab.
// Weights staged in 64KB LDS, rows pair-interleaved so the B fragment of
// V_WMMA_F32_16X16X4_F32 is one ds_load_b64 (the compiler fuses pairs of
// these into ds_load_2addr_b64).
// A fragment layout (ISA 7.12.2, 32-bit A 16x4):
//   lanes 0-15 : row m, (K=k, K=k+1)    lanes 16-31: row m, (K=k+2, K=k+3)
// C/D layout: VGPR v -> M=v (lanes 0-15) / M=v+8 (lanes 16-31), lane&15 -> N.
// The K loop is split into an agg phase (scaled by 1/deg) and an x phase so
// the hot loop is branch-free and software-pipelines cleanly.
// ---------------------------------------------------------------------------
__global__ __launch_bounds__(256) void k_sage_gemm(
    const float* __restrict__ xfeat,  // [N, FEAT] root features
    const float* __restrict__ agg,    // [N, FEAT] summed neighbor features
    const float* __restrict__ deg,    // [N]
    const float* __restrict__ Wl,     // [FEAT, ldW]   (neighbor weight)
    const float* __restrict__ Wr,     // [FEAT, ldW]   (root weight)
    const float* __restrict__ bias,   // [ldW]
    float* __restrict__ out,          // [N, ldOut]
    int nTiles, int ldW, int ldOut) {
  __shared__ float lds[KDIM * COLSB];  // 64 KB, pair-interleaved
  const int colOff = blockIdx.y * COLSB;

  // Stage stacked [Wl ; Wr] slab into LDS: element (r,c) -> lds[((r>>1)*64+c)*2 + (r&1)]
  for (int i = threadIdx.x; i < KDIM * COLSB; i += blockDim.x) {
    int r = i >> 6;
    int c = i & 63;
    float w = (r < FEAT) ? Wl[r * ldW + colOff + c] : Wr[(r - FEAT) * ldW + colOff + c];
    lds[((((r >> 1) << 6) + c) << 1) | (r & 1)] = w;
  }
  __syncthreads();

  const int lane = threadIdx.x & 31;
  const int wave = threadIdx.x >> 5;
  const int wpb = blockDim.x >> 5;
  const int half = lane >> 4;  // which K-pair this lane supplies
  const int m = lane & 15;     // A row / D column within tile

  // bias for this lane's output columns (hoisted; fused into accumulator init)
  float bb[NT];
#pragma unroll
  for (int n = 0; n < NT; ++n) bb[n] = bias[colOff + n * 16 + m];

  // LDS base for this lane's B columns in each tile, in pair-interleaved units
  const float* ldsB = lds + (m << 1);

  for (int tile = blockIdx.x * wpb + wave; tile < nTiles; tile += gridDim.x * wpb) {
    const int row = tile * 16 + m;
    const float inv = __builtin_amdgcn_rcpf(fmaxf(deg[row], 1.0f));  // v_rcp_f32
    const float* aggRow = agg + (size_t)row * FEAT + half * 2;
    const float* xRow = xfeat + (size_t)row * FEAT + half * 2;

    v8f acc[NT];
#pragma unroll
    for (int n = 0; n < NT; ++n) {
#pragma unroll
      for (int j = 0; j < 8; ++j) acc[n][j] = bb[n];
    }

    // ---- phase 1: neighbor-mean part, K = 0..127 (branch-free) ----
#pragma unroll 4
    for (int k = 0; k < FEAT; k += 4) {
      v2f a = *(const v2f*)(aggRow + k);
      a[0] *= inv;
      a[1] *= inv;
      const float* wp = ldsB + (((k >> 1) + half) << 7);  // pair-row base
#pragma unroll
      for (int n = 0; n < NT; ++n) {
        v2f b = *(const v2f*)(wp + (n << 5));
        acc[n] = __builtin_amdgcn_wmma_f32_16x16x4_f32(
            false, a, false, b, (short)0, acc[n], false, false);
      }
    }

    // ---- phase 2: root part, K = 128..255 (branch-free) ----
#pragma unroll 4
    for (int k = 0; k < FEAT; k += 4) {
      v2f a = *(const v2f*)(xRow + k);
      const float* wp = ldsB + ((((k + FEAT) >> 1) + half) << 7);
#pragma unroll
      for (int n = 0; n < NT; ++n) {
        v2f b = *(const v2f*)(wp + (n << 5));
        acc[n] = __builtin_amdgcn_wmma_f32_16x16x4_f32(
            false, a, false, b, (short)0, acc[n], false, false);
      }
    }

    // D tile store
    const int mBase = half ? 8 : 0;
#pragma unroll
    for (int v = 0; v < 8; ++v) {
      float* orow = out + (size_t)(tile * 16 + v + mBase) * ldOut + colOff;
#pragma unroll
      for (int n = 0; n < NT; ++n) orow[n * 16 + m] = acc[n][v];
    }
  }
}

// ---------------------------------------------------------------------------
// LayerNorm + ReLU, in place. One wave32 per node, 4 floats per lane,
// __shfl_xor tree reduction (wave32 on gfx1250).
// ---------------------------------------------------------------------------
__global__ __launch_bounds__(256) void k_ln_relu(
    float* __restrict__ h, const float* __restrict__ g,
    const float* __restrict__ b, int N) {
  int gt = blockIdx.x * blockDim.x + threadIdx.x;
  int node = gt >> 5;
  int lane = gt & 31;
  if (node >= N) return;
  float4 v = *(const float4*)(h + (size_t)node * FEAT + lane * 4);
  float s = v.x + v.y + v.z + v.w;
  float ss = v.x * v.x + v.y * v.y + v.z * v.z + v.w * v.w;
#pragma unroll
  for (int o = 16; o > 0; o >>= 1) {
    s += __shfl_xor(s, o, 32);
    ss += __shfl_xor(ss, o, 32);
  }
  float mu = s * (1.0f / FEAT);
  float var = ss * (1.0f / FEAT) - mu * mu;
  float r = rsqrtf(var + 1e-5f);
  float4 gg = *(const float4*)(g + lane * 4);
  float4 bbv = *(const float4*)(b + lane * 4);
  v.x = fmaxf((v.x - mu) * r * gg.x + bbv.x, 0.0f);
  v.y = fmaxf((v.y - mu) * r * gg.y + bbv.y, 0.0f);
  v.z = fmaxf((v.z - mu) * r * gg.z + bbv.z, 0.0f);
  v.w = fmaxf((v.w - mu) * r * gg.w + bbv.w, 0.0f);
  *(float4*)(h + (size_t)node * FEAT + lane * 4) = v;
}

// ---------------------------------------------------------------------------
// Row-wise L2 normalize [N,64] output, in place. One wave per node.
// ---------------------------------------------------------------------------
__global__ __launch_bounds__(256) void k_l2(float* __restrict__ out, int N) {
  int gt = blockIdx.x * blockDim.x + threadIdx.x;
  int node = gt >> 5;
  int lane = gt & 31;
  if (node >= N) return;
  float2 v = *(const float2*)(out + (size_t)node * 64 + lane * 2);
  float ss = v.x * v.x + v.y * v.y;
#pragma unroll
  for (int o = 16; o > 0; o >>= 1) ss += __shfl_xor(ss, o, 32);
  float inv = 1.0f / fmaxf(sqrtf(ss), 1e-12f);
  v.x *= inv;
  v.y *= inv;
  *(float2*)(out + (size_t)node * 64 + lane * 2) = v;
}

// ---------------------------------------------------------------------------
// Host launcher. Inputs (setup_inputs order):
//   0:x [N,128] f32   1:edge_index [2,E] i32   2:W1l [128,128]  3:b1l [128]
//   4:W1r [128,128]   5:ln_g [128]  6:ln_b [128]
//   7:W2l [128,64]    8:b2l [64]    9:W2r [128,64]
// Workspace: deg [N] | agg [N,128] | h [N,128]  (~103 MB)
// ---------------------------------------------------------------------------
extern "C" void kernel_launch(void* const* d_in, const int* in_sizes, int n_in,
                              void* d_out, int out_size, void* d_ws, size_t ws_size,
                              hipStream_t stream) {
  const float* x   = (const float*)d_in[0];
  const int*   ei  = (const int*)d_in[1];
  const float* W1l = (const float*)d_in[2];
  const float* b1l = (const float*)d_in[3];
  const float* W1r = (const float*)d_in[4];
  const float* lng = (const float*)d_in[5];
  const float* lnb = (const float*)d_in[6];
  const float* W2l = (const float*)d_in[7];
  const float* b2l = (const float*)d_in[8];
  const float* W2r = (const float*)d_in[9];

  const int N = in_sizes[0] / FEAT;
  const int E = in_sizes[1] / 2;
  const int* src = ei;
  const int* dst = ei + E;

  float* deg = (float*)d_ws;
  float* agg = deg + N;
  float* h   = agg + (size_t)N * FEAT;
  float* outp = (float*)d_out;

  const int nTiles = N / 16;
  const int gemmBlocksX = (nTiles + 7) / 8;       // 8 waves (tiles) per block
  const int edgeWaveBlocks = (E + 7) / 8;         // 8 edge-waves per block
  const int nodeWaveBlocks = (N + 7) / 8;         // 8 node-waves per block

  // 1) zero deg + agg
  k_zero<<<4096, 256, 0, stream>>>(deg, (long)N * (1 + FEAT));
  // 2) degrees
  k_deg<<<(E + 255) / 256, 256, 0, stream>>>(dst, deg, E);
  // 3) scatter x into agg
  k_scatter<<<edgeWaveBlocks, 256, 0, stream>>>(x, src, dst, agg, E);
  // 4) layer 1: h = (agg/deg)@W1l + b1l + x@W1r   (two 64-col slabs)
  {
    dim3 g(gemmBlocksX, 2);
    k_sage_gemm<<<g, 256, 0, stream>>>(x, agg, deg, W1l, W1r, b1l, h,
                                       nTiles, 128, 128);
  }
  // 5) LayerNorm + ReLU in place
  k_ln_relu<<<nodeWaveBlocks, 256, 0, stream>>>(h, lng, lnb, N);
  // 6) re-zero agg
  k_zero<<<4096, 256, 0, stream>>>(agg, (long)N * FEAT);
  // 7) scatter h into agg
  k_scatter<<<edgeWaveBlocks, 256, 0, stream>>>(h, src, dst, agg, E);
  // 8) layer 2: out = (agg/deg)@W2l + b2l + h@W2r   (single 64-col slab)
  {
    dim3 g(gemmBlocksX, 1);
    k_sage_gemm<<<g, 256, 0, stream>>>(h, agg, deg, W2l, W2r, b2l, outp,
                                       nTiles, 64, 64);
  }
  // 9) row-wise L2 normalize
  k_l2<<<nodeWaveBlocks, 256, 0, stream>>>(outp, N);
}